// QuantumNeuralNetwork_90804198572547
// MI455X (gfx1250) — compile-verified
//
#include <hip/hip_runtime.h>

typedef __attribute__((ext_vector_type(2))) float v2f;
typedef __attribute__((ext_vector_type(8))) float v8f;
typedef __attribute__((ext_vector_type(2))) int   i32x2;

#define NQ 7
#define DIM 128
#define NL 2
#define SSTRIDE 132   // padded LDS row stride (floats): conflict-free b64 A loads

// CDNA5 async global->LDS staging (ASYNCcnt-tracked), with safe fallback.
#if __has_builtin(__builtin_amdgcn_global_load_async_to_lds_b64) && \
    __has_builtin(__builtin_amdgcn_s_wait_asynccnt)
#define QNN_ASYNC 1
typedef __attribute__((address_space(1))) i32x2 as1_i32x2;
typedef __attribute__((address_space(3))) i32x2 as3_i32x2;
#endif

// ---------------------------------------------------------------------------
// Kernel 1: build the fixed 128x128 complex unitary U for the 2 weight layers.
// grid: 8 blocks x 128 threads; block b evolves identity columns [16b,16b+16).
// Output row-major U[j][k] (j = output basis index, k = input column).
// ---------------------------------------------------------------------------
__global__ __launch_bounds__(128)
void build_unitary(const float* __restrict__ w,
                   float* __restrict__ Ure, float* __restrict__ Uim) {
    __shared__ float re[16 * DIM];
    __shared__ float im[16 * DIM];
    const int tid = threadIdx.x;          // 0..127
    const int colbase = blockIdx.x * 16;

    for (int t = tid; t < 16 * DIM; t += 128) {
        int c = t >> 7, j = t & 127;
        re[t] = (j == colbase + c) ? 1.f : 0.f;
        im[t] = 0.f;
    }
    __syncthreads();

    for (int l = 0; l < NL; ++l) {
        for (int wi = 0; wi < NQ; ++wi) {
            {   // RY(weights[l][wi][0]) on wire wi
                float th = 0.5f * w[(l * NQ + wi) * 2 + 0];
                float c = cosf(th), s = sinf(th);
                int b = 1 << (6 - wi);
                for (int t = tid; t < 16 * 64; t += 128) {
                    int cc = t >> 6, p = t & 63;
                    int lo = ((p & ~(b - 1)) << 1) | (p & (b - 1));
                    int j0 = cc * DIM + lo, j1 = j0 + b;
                    float r0 = re[j0], r1 = re[j1];
                    float i0 = im[j0], i1 = im[j1];
                    re[j0] = c * r0 - s * r1;  re[j1] = s * r0 + c * r1;
                    im[j0] = c * i0 - s * i1;  im[j1] = s * i0 + c * i1;
                }
            }
            __syncthreads();
            {   // RZ(weights[l][wi][1]) on wire wi
                float th = 0.5f * w[(l * NQ + wi) * 2 + 1];
                float c = cosf(th), s = sinf(th);
                for (int t = tid; t < 16 * DIM; t += 128) {
                    int j = t & 127;
                    int bit = (j >> (6 - wi)) & 1;
                    float r = re[t], m = im[t];
                    if (bit == 0) { re[t] = c * r + s * m; im[t] = c * m - s * r; }
                    else          { re[t] = c * r - s * m; im[t] = c * m + s * r; }
                }
            }
            __syncthreads();
        }
        for (int e = 0; e < NQ; ++e) {   // CNOT ring (0,1)..(5,6),(6,0)
            int cw = e, tw = (e + 1) % NQ;
            int cb = 1 << (6 - cw), tb = 1 << (6 - tw);
            for (int t = tid; t < 16 * DIM; t += 128) {
                int j = t & 127;
                if ((j & cb) && !(j & tb)) {
                    int t2 = t + tb;
                    float r = re[t], m = im[t];
                    re[t] = re[t2]; im[t] = im[t2];
                    re[t2] = r;     im[t2] = m;
                }
            }
            __syncthreads();
        }
    }
    for (int t = tid; t < 16 * DIM; t += 128) {
        int c = t >> 7, j = t & 127;
        Ure[j * DIM + colbase + c] = re[t];
        Uim[j * DIM + colbase + c] = im[t];
    }
}

// ---------------------------------------------------------------------------
// Kernel 2: 64 batch rows per block (256 threads, 8 waves).
//  LDS: S[64][132] (later reused for probs) + 3 triple-buffered U k-panels.
//  Pipeline: stage panel k+2 (async->LDS) while WMMA-ing panel k.
// ---------------------------------------------------------------------------
__global__ __launch_bounds__(256)
void qnn_main(const float* __restrict__ x,
              const float* __restrict__ Ure, const float* __restrict__ Uim,
              float* __restrict__ out) {
    extern __shared__ float dyn[];
    float* S    = dyn;                    // 64*132 floats
    float* Bpan = dyn + 64 * SSTRIDE;     // 3 bufs * (512 re + 512 im) floats

    const int tid  = threadIdx.x;
    const int lane = tid & 31;
    const int wave = tid >> 5;            // 0..7
    const int rowbase = blockIdx.x * 64;

    // panel staging: copy U[j][kk..kk+3] (re & im) into panel buffer `buf`
    auto stage = [&](int kk, int buf) {
        int j    = tid >> 1;              // 0..127
        int half = (tid & 1) * 2;         // 0 or 2
        float* dst = Bpan + buf * 1024 + j * 4 + half;
#ifdef QNN_ASYNC
        __builtin_amdgcn_global_load_async_to_lds_b64(
            (as1_i32x2*)(Ure + j * DIM + kk + half), (as3_i32x2*)dst, 0, 0);
        __builtin_amdgcn_global_load_async_to_lds_b64(
            (as1_i32x2*)(Uim + j * DIM + kk + half), (as3_i32x2*)(dst + 512), 0, 0);
#else
        v2f r = *(const v2f*)(Ure + j * DIM + kk + half);
        v2f m = *(const v2f*)(Uim + j * DIM + kk + half);
        *(v2f*)dst = r;
        *(v2f*)(dst + 512) = m;
#endif
    };

    // ---- Phase 0: product states S[64][128] into LDS; prime panel pipeline
    stage(0, 0);
    stage(4, 1);
    {
        int r = tid >> 2;                 // 4 threads per row
        int kbase = (tid & 3) * 32;
        const float* xr = x + (rowbase + r) * NQ;
        float cs[NQ], sn[NQ];
#pragma unroll
        for (int i = 0; i < NQ; ++i) {
            float th = 0.5f * xr[i];
            cs[i] = __cosf(th);
            sn[i] = __sinf(th);
        }
        for (int kk = 0; kk < 32; ++kk) {
            int k = kbase + kk;
            float a = 1.f;
#pragma unroll
            for (int i = 0; i < NQ; ++i)
                a *= ((k >> (6 - i)) & 1) ? sn[i] : cs[i];
            S[r * SSTRIDE + k] = a;
        }
    }
#ifdef QNN_ASYNC
    __builtin_amdgcn_s_wait_asynccnt(2);  // panel 0 complete (panel 1 in flight)
#endif
    __syncthreads();

    // ---- Phase 1: final_re/final_im tiles via WMMA, panel-pipelined ----
    const int mt    = wave >> 1;          // 0..3  (M tile)
    const int ntb   = (wave & 1) * 4;     // 0 or 4 (base N tile)
    const int n     = lane & 15;
    const int khalf = (lane >> 4) * 2;

    v8f acc_re[4] = {{0.f}, {0.f}, {0.f}, {0.f}};
    v8f acc_im[4] = {{0.f}, {0.f}, {0.f}, {0.f}};
    const float* sA = S + (mt * 16 + n) * SSTRIDE;

    int cur = 0, nxt = 2;                 // buffer being consumed / staged
    for (int i = 0; i < 32; ++i) {
        if (i + 2 < 32) stage((i + 2) * 4, nxt);

        const float* bb = Bpan + cur * 1024;
        int k0 = i * 4 + khalf;
        v2f a = *(const v2f*)(sA + k0);
#pragma unroll
        for (int t = 0; t < 4; ++t) {
            int boff = ((ntb + t) * 16 + n) * 4 + khalf;
            v2f br = *(const v2f*)(bb + boff);
            v2f bi = *(const v2f*)(bb + 512 + boff);
            acc_re[t] = __builtin_amdgcn_wmma_f32_16x16x4_f32(
                false, a, false, br, (short)0, acc_re[t], false, false);
            acc_im[t] = __builtin_amdgcn_wmma_f32_16x16x4_f32(
                false, a, false, bi, (short)0, acc_im[t], false, false);
        }
#ifdef QNN_ASYNC
        if (i + 2 < 32) __builtin_amdgcn_s_wait_asynccnt(2);  // next panel ready
        else            __builtin_amdgcn_s_wait_asynccnt(0);  // drain
#endif
        __syncthreads();
        cur = (cur == 2) ? 0 : cur + 1;
        nxt = (nxt == 2) ? 0 : nxt + 1;
    }

    // ---- Phase 2a: probs = re^2 + im^2 into LDS (S region reused) ----
    {
        int hi = (lane >> 4) * 8;         // D layout: M = vgpr + 8*(lane/16)
#pragma unroll
        for (int t = 0; t < 4; ++t) {
            int j = (ntb + t) * 16 + n;
#pragma unroll
            for (int r = 0; r < 8; ++r) {
                float pr = acc_re[t][r], pi = acc_im[t][r];
                S[(mt * 16 + r + hi) * SSTRIDE + j] = pr * pr + pi * pi;
            }
        }
    }
    __syncthreads();

    // ---- Phase 2b: out = probs @ Z^T via WMMA (waves 0..3) ----
    if (wave < 4) {
        v8f acc = {0.f};
        const float* pA = S + (wave * 16 + n) * SSTRIDE;
        int sh = (6 - n) & 31;            // valid only when n < 7
        for (int kk = 0; kk < DIM; kk += 4) {
            int k0 = kk + khalf;
            v2f a = *(const v2f*)(pA + k0);
            v2f b;
            b.x = (n < 7) ? ((((k0    ) >> sh) & 1) ? -1.f : 1.f) : 0.f;
            b.y = (n < 7) ? ((((k0 + 1) >> sh) & 1) ? -1.f : 1.f) : 0.f;
            acc = __builtin_amdgcn_wmma_f32_16x16x4_f32(
                false, a, false, b, (short)0, acc, false, false);
        }
        if (n < 7) {
            int hi = (lane >> 4) * 8;
#pragma unroll
            for (int r = 0; r < 8; ++r) {
                int row = rowbase + wave * 16 + r + hi;
                out[row * NQ + n] = acc[r];
            }
        }
    }
}

// ---------------------------------------------------------------------------
extern "C" void kernel_launch(void* const* d_in, const int* in_sizes, int n_in,
                              void* d_out, int out_size, void* d_ws, size_t ws_size,
                              hipStream_t stream) {
    const float* x = (const float*)d_in[0];      // (65536, 7) f32
    const float* w = (const float*)d_in[1];      // (2, 7, 2) f32
    float* Ure = (float*)d_ws;                   // 128*128 f32
    float* Uim = Ure + DIM * DIM;                // 128*128 f32

    build_unitary<<<8, 128, 0, stream>>>(w, Ure, Uim);

    const size_t lds_bytes = (64 * SSTRIDE + 3 * 1024) * sizeof(float); // 46080
    qnn_main<<<65536 / 64, 256, lds_bytes, stream>>>(x, Ure, Uim, (float*)d_out);
}